// GSNN_Decoder_GAT_80633716015245
// MI455X (gfx1250) — compile-verified
//
#include <hip/hip_runtime.h>
#include <math.h>

typedef __attribute__((ext_vector_type(2))) float v2f;
typedef __attribute__((ext_vector_type(8))) float v8f;

#define NNODES 50000
#define NEDGES 800000

__device__ __forceinline__ float leaky(float x) { return x > 0.f ? x : 0.2f * x; }

// ---------------------------------------------------------------------------
// WMMA f32 GEMM: C[M,Nc] = act(A[M,K] @ B[K,Nc] + bias). Row-major A and B.
// One wave computes CTW adjacent 16x16 output tiles (one row tile, CTW col
// tiles) with V_WMMA_F32_16X16X4_F32: A fragment reused CTW times per k-step,
// CTW independent accumulator chains keep the matrix pipe busy.
// M, Nc multiples of 16; Nc multiple of 16*CTW; K multiple of 4.
// ACT: 0 = none, 1 = relu.
// ---------------------------------------------------------------------------
template <int ACT, int CTW>
__global__ void wmma_gemm(const float* __restrict__ A, const float* __restrict__ B,
                          const float* __restrict__ bias, float* __restrict__ C,
                          int M, int Nc, int K) {
  const int lane = threadIdx.x & 31;
  const int wave = blockIdx.x * (blockDim.x >> 5) + (threadIdx.x >> 5);
  const int colGroups = Nc >> (4 + (CTW == 4 ? 2 : 1));
  const int totalWaves = (M >> 4) * colGroups;
  if (wave >= totalWaves) return;  // wave-uniform: EXEC all ones for WMMA
  const int rt = wave / colGroups;
  const int cg = wave - rt * colGroups;
  const int l15 = lane & 15;
  const int khalf = (lane >> 4) << 1;  // lanes 0-15 -> K 0,1 ; lanes 16-31 -> K 2,3
  const int m = (rt << 4) + l15;       // A fragment row (M on lanes)
  const int n0 = (cg * CTW) << 4;      // first col of this wave's tile group
  const float* arow = A + (long)m * K + khalf;
  const float* bbase = B + (long)khalf * Nc + n0 + l15;
  v8f acc[CTW] = {};
  for (int k = 0; k < K; k += 4) {
    v2f a;
    a.x = arow[k];
    a.y = arow[k + 1];
#pragma unroll
    for (int t = 0; t < CTW; ++t) {
      v2f b;
      b.x = bbase[(long)k * Nc + t * 16];
      b.y = bbase[(long)(k + 1) * Nc + t * 16];
      acc[t] = __builtin_amdgcn_wmma_f32_16x16x4_f32(false, a, false, b, (short)0,
                                                     acc[t], false, false);
    }
  }
  // D layout: VGPR i holds row rt*16+i (lanes 0-15) and rt*16+i+8 (lanes 16-31)
  const int rbase = (rt << 4) + ((lane >> 4) << 3);
#pragma unroll
  for (int t = 0; t < CTW; ++t) {
    const int col = n0 + t * 16 + l15;
    const float bv = bias ? bias[col] : 0.f;
#pragma unroll
    for (int i = 0; i < 8; ++i) {
      float v = acc[t][i] + bv;
      if (ACT == 1) v = fmaxf(v, 0.f);
      C[(long)(rbase + i) * Nc + col] = v;
    }
  }
}

// ---------------------------------------------------------------------------
// h1[n,0:64] = h64[n], h1[n,64:128] = z ; then L2-normalize each row (+1e-6).
// One wave per node, 4 elements per lane.
// ---------------------------------------------------------------------------
__global__ void concat_norm(const float* __restrict__ h64, const float* __restrict__ z,
                            float* __restrict__ h1) {
  const int lane = threadIdx.x & 31;
  const int node = blockIdx.x * (blockDim.x >> 5) + (threadIdx.x >> 5);
  if (node >= NNODES) return;
  const int base = lane << 2;  // 0..124
  float v[4];
#pragma unroll
  for (int j = 0; j < 4; ++j) {
    const int c = base + j;
    v[j] = (c < 64) ? h64[(long)node * 64 + c] : z[c - 64];
  }
  float s = v[0] * v[0] + v[1] * v[1] + v[2] * v[2] + v[3] * v[3];
#pragma unroll
  for (int off = 16; off > 0; off >>= 1) s += __shfl_xor(s, off, 32);
  const float inv = 1.f / (sqrtf(s) + 1e-6f);
#pragma unroll
  for (int j = 0; j < 4; ++j) h1[(long)node * 128 + base + j] = v[j] * inv;
}

// ---------------------------------------------------------------------------
// el[w] = dot(feat[w*D:...], al[h*D:...]), same for er. w = node*H + h.
// One wave per (node, head).
// ---------------------------------------------------------------------------
__global__ void attn_logits(const float* __restrict__ feat, const float* __restrict__ al,
                            const float* __restrict__ ar, float* __restrict__ el,
                            float* __restrict__ er, int H, int D) {
  const int lane = threadIdx.x & 31;
  const int w = blockIdx.x * (blockDim.x >> 5) + (threadIdx.x >> 5);
  if (w >= NNODES * H) return;
  const float* f = feat + (long)w * D;
  const int h = w % H;
  const float* a = al + h * D;
  const float* r = ar + h * D;
  const int per = D >> 5;
  float sl = 0.f, sr = 0.f;
  for (int j = 0; j < per; ++j) {
    const float x = f[lane * per + j];
    sl += x * a[lane * per + j];
    sr += x * r[lane * per + j];
  }
#pragma unroll
  for (int off = 16; off > 0; off >>= 1) {
    sl += __shfl_xor(sl, off, 32);
    sr += __shfl_xor(sr, off, 32);
  }
  if (lane == 0) {
    el[w] = sl;
    er[w] = sr;
  }
}

__global__ void init_softmax(float* __restrict__ emax, float* __restrict__ den, int n) {
  const int i = blockIdx.x * blockDim.x + threadIdx.x;
  if (i < n) {
    emax[i] = -3.0e38f;
    den[i] = 0.f;
  }
}

__global__ void init_rst_bias(float* __restrict__ rst, const float* __restrict__ bias,
                              long total, int cmask) {
  const long i = (long)blockIdx.x * blockDim.x + threadIdx.x;
  if (i < total) rst[i] = bias[i & cmask];
}

__global__ void init_rst_bias_res(float* __restrict__ rst, const float* __restrict__ bias,
                                  const float* __restrict__ res, long total, int cmask) {
  const long i = (long)blockIdx.x * blockDim.x + threadIdx.x;
  if (i < total) rst[i] = bias[i & cmask] + res[i];
}

// ---------------------------------------------------------------------------
// Edge softmax passes: one thread per edge, both heads handled in-thread so
// src/dst indices are loaded once. el/er/emax/den stay L2-resident.
// ---------------------------------------------------------------------------
template <int H>
__global__ void edge_max(const int* __restrict__ src, const int* __restrict__ dst,
                         const float* __restrict__ el, const float* __restrict__ er,
                         float* __restrict__ emax) {
  const int e = blockIdx.x * blockDim.x + threadIdx.x;
  if (e >= NEDGES) return;
  const int s = src[e], d = dst[e];
#pragma unroll
  for (int h = 0; h < H; ++h) {
    const float v = leaky(el[s * H + h] + er[d * H + h]);
    atomicMax(&emax[d * H + h], v);
  }
}

// Pass 2: accumulate denominator AND cache each edge's softmax numerator in
// exbuf so the heavy scatter pass does a single load instead of re-gathering
// el/er/emax and recomputing expf.
template <int H>
__global__ void edge_expsum(const int* __restrict__ src, const int* __restrict__ dst,
                            const float* __restrict__ el, const float* __restrict__ er,
                            const float* __restrict__ emax, float* __restrict__ den,
                            float* __restrict__ exbuf) {
  const int e = blockIdx.x * blockDim.x + threadIdx.x;
  if (e >= NEDGES) return;
  const int s = src[e], d = dst[e];
#pragma unroll
  for (int h = 0; h < H; ++h) {
    const float v = leaky(el[s * H + h] + er[d * H + h]);
    const float ex = expf(v - emax[d * H + h]);
    exbuf[e * H + h] = ex;
    atomicAdd(&den[d * H + h], ex);
  }
}

// Scatter for H=2, D=128 (256 cols): one wave per edge; lanes 0-15 head0,
// lanes 16-31 head1, 8 contiguous floats per lane (2x float4 gather).
__global__ void edge_scatter256(const int* __restrict__ src, const int* __restrict__ dst,
                                const float* __restrict__ feat,
                                const float* __restrict__ exbuf,
                                const float* __restrict__ den, float* __restrict__ rst) {
  const int lane = threadIdx.x & 31;
  const int e = blockIdx.x * (blockDim.x >> 5) + (threadIdx.x >> 5);
  if (e >= NEDGES) return;
  const int s = src[e], d = dst[e];
  const int h = lane >> 4;
  const float alpha = exbuf[e * 2 + h] / den[d * 2 + h];
  const int cb = ((lane & 15) << 3) + (h << 7);  // head0: 0..127, head1: 128..255
  const float4 f0 = *(const float4*)(feat + (long)s * 256 + cb);
  const float4 f1 = *(const float4*)(feat + (long)s * 256 + cb + 4);
  float* o = rst + (long)d * 256 + cb;
  atomicAdd(&o[0], f0.x * alpha);
  atomicAdd(&o[1], f0.y * alpha);
  atomicAdd(&o[2], f0.z * alpha);
  atomicAdd(&o[3], f0.w * alpha);
  atomicAdd(&o[4], f1.x * alpha);
  atomicAdd(&o[5], f1.y * alpha);
  atomicAdd(&o[6], f1.z * alpha);
  atomicAdd(&o[7], f1.w * alpha);
}

// Scatter for final layer: H=1, 32 cols; one wave per edge, one class per lane.
__global__ void edge_scatter32(const int* __restrict__ src, const int* __restrict__ dst,
                               const float* __restrict__ feat,
                               const float* __restrict__ exbuf,
                               const float* __restrict__ den, float* __restrict__ out) {
  const int lane = threadIdx.x & 31;
  const int e = blockIdx.x * (blockDim.x >> 5) + (threadIdx.x >> 5);
  if (e >= NEDGES) return;
  const int s = src[e], d = dst[e];
  const float alpha = exbuf[e] / den[d];
  atomicAdd(&out[(long)d * 32 + lane], feat[(long)s * 32 + lane] * alpha);
}

__global__ void elu_inplace(float* __restrict__ x, long n) {
  const long i = (long)blockIdx.x * blockDim.x + threadIdx.x;
  if (i < n) {
    const float v = x[i];
    x[i] = v > 0.f ? v : expf(v) - 1.f;
  }
}

// ---------------------------------------------------------------------------
extern "C" void kernel_launch(void* const* d_in, const int* in_sizes, int n_in,
                              void* d_out, int out_size, void* d_ws, size_t ws_size,
                              hipStream_t stream) {
  const float* inputs = (const float*)d_in[0];
  const float* z      = (const float*)d_in[1];
  const int*   src    = (const int*)d_in[2];
  const int*   dst    = (const int*)d_in[3];
  const float* Wx     = (const float*)d_in[4];
  const float* bx     = (const float*)d_in[5];
  const float* W0     = (const float*)d_in[6];
  const float* al0    = (const float*)d_in[7];
  const float* ar0    = (const float*)d_in[8];
  const float* b0     = (const float*)d_in[9];
  const float* W1     = (const float*)d_in[10];
  const float* al1    = (const float*)d_in[11];
  const float* ar1    = (const float*)d_in[12];
  const float* b1     = (const float*)d_in[13];
  const float* Wf     = (const float*)d_in[14];
  const float* alf    = (const float*)d_in[15];
  const float* arf    = (const float*)d_in[16];
  const float* bf     = (const float*)d_in[17];
  const float* resWf  = (const float*)d_in[18];
  float* out = (float*)d_out;

  const long N = NNODES;
  float* ws    = (float*)d_ws;
  float* BUF0  = ws;                // N*256
  float* BUF1  = ws + N * 256;      // N*256
  float* BUF2  = ws + N * 256 * 2;  // N*256
  float* el    = ws + N * 256 * 3;  // N*2
  float* er    = el + N * 2;
  float* emax  = er + N * 2;
  float* den   = emax + N * 2;
  float* exbuf = den + N * 2;       // E*2
  (void)in_sizes; (void)n_in; (void)out_size; (void)ws_size;

  const int TPB = 256;  // 8 waves
  auto cdiv = [](long a, long b) { return (int)((a + b - 1) / b); };

  // ---- x_to_h: h64 = relu(inputs @ Wx + bx)  [N,64] -> BUF1 ----
  wmma_gemm<1, 4><<<cdiv(N / 16, 8), TPB, 0, stream>>>(inputs, Wx, bx, BUF1,
                                                       NNODES, 64, 256);
  // ---- h1 = normalize(concat(h64, z))  [N,128] -> BUF0 ----
  concat_norm<<<cdiv(N, 8), TPB, 0, stream>>>(BUF1, z, BUF0);

  // ================= GAT layer 0 (H=2, D=128) =================
  wmma_gemm<0, 4><<<cdiv((N / 16) * 4, 8), TPB, 0, stream>>>(BUF0, W0, nullptr, BUF1,
                                                             NNODES, 256, 128);
  attn_logits<<<cdiv(N * 2, 8), TPB, 0, stream>>>(BUF1, al0, ar0, el, er, 2, 128);
  init_softmax<<<cdiv(N * 2, TPB), TPB, 0, stream>>>(emax, den, NNODES * 2);
  init_rst_bias<<<cdiv(N * 256, TPB), TPB, 0, stream>>>(BUF2, b0, N * 256, 255);
  edge_max<2><<<cdiv(NEDGES, TPB), TPB, 0, stream>>>(src, dst, el, er, emax);
  edge_expsum<2><<<cdiv(NEDGES, TPB), TPB, 0, stream>>>(src, dst, el, er, emax, den, exbuf);
  edge_scatter256<<<cdiv(NEDGES, 8), TPB, 0, stream>>>(src, dst, BUF1, exbuf, den, BUF2);
  elu_inplace<<<cdiv(N * 256, TPB), TPB, 0, stream>>>(BUF2, N * 256);  // h2 in BUF2

  // ================= GAT layer 1 (H=2, D=128, identity residual) =================
  wmma_gemm<0, 4><<<cdiv((N / 16) * 4, 8), TPB, 0, stream>>>(BUF2, W1, nullptr, BUF0,
                                                             NNODES, 256, 256);
  attn_logits<<<cdiv(N * 2, 8), TPB, 0, stream>>>(BUF0, al1, ar1, el, er, 2, 128);
  init_softmax<<<cdiv(N * 2, TPB), TPB, 0, stream>>>(emax, den, NNODES * 2);
  init_rst_bias_res<<<cdiv(N * 256, TPB), TPB, 0, stream>>>(BUF1, b1, BUF2, N * 256, 255);
  edge_max<2><<<cdiv(NEDGES, TPB), TPB, 0, stream>>>(src, dst, el, er, emax);
  edge_expsum<2><<<cdiv(NEDGES, TPB), TPB, 0, stream>>>(src, dst, el, er, emax, den, exbuf);
  edge_scatter256<<<cdiv(NEDGES, 8), TPB, 0, stream>>>(src, dst, BUF0, exbuf, den, BUF1);
  elu_inplace<<<cdiv(N * 256, TPB), TPB, 0, stream>>>(BUF1, N * 256);  // h3 in BUF1

  // ================= Final GAT (H=1, C=32, linear residual) =================
  wmma_gemm<0, 2><<<cdiv(N / 16, 8), TPB, 0, stream>>>(BUF1, Wf, nullptr, BUF2,
                                                       NNODES, 32, 256);
  attn_logits<<<cdiv(N, 8), TPB, 0, stream>>>(BUF2, alf, arf, el, er, 1, 32);
  init_softmax<<<cdiv(N, TPB), TPB, 0, stream>>>(emax, den, NNODES);
  // out = h3 @ resWf + bf (then edge messages accumulate on top)
  wmma_gemm<0, 2><<<cdiv(N / 16, 8), TPB, 0, stream>>>(BUF1, resWf, bf, out,
                                                       NNODES, 32, 256);
  edge_max<1><<<cdiv(NEDGES, TPB), TPB, 0, stream>>>(src, dst, el, er, emax);
  edge_expsum<1><<<cdiv(NEDGES, TPB), TPB, 0, stream>>>(src, dst, el, er, emax, den, exbuf);
  edge_scatter32<<<cdiv(NEDGES, 8), TPB, 0, stream>>>(src, dst, BUF2, exbuf, den, out);
}